// AttentionBlock3D_39582418600129
// MI455X (gfx1250) — compile-verified
//
#include <hip/hip_runtime.h>
#include <math.h>

#define B_   2
#define C_   128
#define S_   13824          // 24*24*24
#define G_   8
#define CPG  16
#define NH_  4
#define HD_  32
#define HP_  12
#define SP_  1728           // 12*12*12
#define SCALE_ 0.17677669529663687f   // 1/sqrt(32)

typedef float v2f __attribute__((ext_vector_type(2)));
typedef float v8f __attribute__((ext_vector_type(8)));
typedef unsigned int u32x4 __attribute__((ext_vector_type(4)));
typedef int i32x4 __attribute__((ext_vector_type(4)));
typedef int i32x8 __attribute__((ext_vector_type(8)));

#if __has_builtin(__builtin_amdgcn_tensor_load_to_lds)
#define HAVE_TDM 1
#else
#define HAVE_TDM 0
#endif

__device__ __forceinline__ v8f wmma_step(v8f acc, v2f a, v2f b) {
  // D = A(16x4 f32) * B(4x16 f32) + C(16x16 f32)
  return __builtin_amdgcn_wmma_f32_16x16x4_f32(false, a, false, b, (short)0, acc, false, false);
}

#if HAVE_TDM
// TDM: load a 16x128 f32 tile (16 contiguous rows of W) into LDS, inserting a
// 1-dword pad every 128 dwords so the LDS row stride is 129 (bank-conflict-free
// A-fragment ds loads: lane stride 129 mod 64 == 1).
__device__ __forceinline__ void tdm_load_w16x128(const float* gsrc, unsigned lds_off) {
  unsigned long long ga = (unsigned long long)(size_t)gsrc;
  u32x4 g0 = { 1u,                                  // count=1, no gather
               lds_off,                             // lds_addr
               (unsigned)(ga & 0xffffffffu),        // global_addr[31:0]
               (unsigned)((ga >> 32) & 0x01ffffffu) | (2u << 30) };  // ga[56:32] | type=2
  i32x8 g1;
  g1[0] = (2 << 16) | (1 << 20) | (6 << 22);        // data_size=4B, pad_enable, interval=128dw, amount=1dw
  g1[1] = (int)(128u << 16);                        // tensor_dim0 = 128
  g1[2] = (int)(16u << 16);                         // tensor_dim1 = 16
  g1[3] = (int)(128u << 16);                        // tile_dim0 = 128
  g1[4] = 16;                                       // tile_dim1 = 16, tile_dim2 = 0
  g1[5] = 128;                                      // tensor_dim0_stride = 128
  g1[6] = 0;
  g1[7] = 0;
  i32x4 z4 = {0, 0, 0, 0};
#if defined(__clang_major__) && __clang_major__ >= 23
  i32x8 z8 = {0, 0, 0, 0, 0, 0, 0, 0};
  __builtin_amdgcn_tensor_load_to_lds(g0, g1, z4, z4, z8, 0);
#else
  __builtin_amdgcn_tensor_load_to_lds(g0, g1, z4, z4, 0);
#endif
}
#endif

// ---------------- Kernel 1: GroupNorm statistics ----------------
__global__ void __launch_bounds__(256) gn_stats(const float* __restrict__ x, float* __restrict__ stats) {
  int bg = blockIdx.x;               // B_*G_ blocks
  const float* p = x + (size_t)bg * CPG * S_;   // groups are channel-contiguous
  const int n = CPG * S_;            // 221184
  float s = 0.f, s2 = 0.f;
  for (int i = threadIdx.x; i < n; i += 256) { float v = p[i]; s += v; s2 += v * v; }
  __shared__ float rs[256], rs2[256];
  rs[threadIdx.x] = s; rs2[threadIdx.x] = s2;
  __syncthreads();
  for (int off = 128; off > 0; off >>= 1) {
    if (threadIdx.x < off) { rs[threadIdx.x] += rs[threadIdx.x + off]; rs2[threadIdx.x] += rs2[threadIdx.x + off]; }
    __syncthreads();
  }
  if (threadIdx.x == 0) {
    float mu = rs[0] / (float)n;
    float var = rs2[0] / (float)n - mu * mu;
    stats[bg * 2 + 0] = mu;
    stats[bg * 2 + 1] = rsqrtf(var + 1e-5f);
  }
}

// ---------------- Kernel 2: GroupNorm apply ----------------
__global__ void __launch_bounds__(256) gn_apply(const float* __restrict__ x, const float* __restrict__ w,
                                                const float* __restrict__ bb, const float* __restrict__ stats,
                                                float* __restrict__ xn) {
  size_t i = (size_t)blockIdx.x * 256 + threadIdx.x;      // exactly B_*C_*S_ threads
  int c = (int)((i / S_) % C_);
  int b = (int)(i / ((size_t)C_ * S_));
  int g = c / CPG;
  float mu = stats[(b * G_ + g) * 2 + 0];
  float rs = stats[(b * G_ + g) * 2 + 1];
  xn[i] = (x[i] - mu) * rs * w[c] + bb[c];
}

// ---------------- Kernels 3/6: fp32 WMMA GEMM  Out[b,m,s] = W[m,:]*X[b,:,s] + bias (+resid) ----------------
// 8 waves/block share one 16-row W tile, staged once into LDS by the TDM.
template <int M_, bool RES>
__global__ void __launch_bounds__(256) gemm_wmma(const float* __restrict__ X, const float* __restrict__ Wt,
                                                 const float* __restrict__ bias, const float* __restrict__ resid,
                                                 float* __restrict__ Out) {
  const int Mt = M_ / 16, Nt = S_ / 16;
  __shared__ float Wlds[16 * 129];                 // padded row stride 129
  // m0 / b are block-uniform: Nt (=864) and Mt*Nt are multiples of 8
  int bw0 = blockIdx.x * 8;
  int b   = bw0 / (Mt * Nt);
  int m0  = ((bw0 % (Mt * Nt)) / Nt) * 16;
#if HAVE_TDM
  if (threadIdx.x < 32) {
    tdm_load_w16x128(Wt + (size_t)m0 * C_, (unsigned)(size_t)(void*)Wlds);
    __builtin_amdgcn_s_wait_tensorcnt(0);
  }
#else
  for (int i = threadIdx.x; i < 16 * C_; i += 256)
    Wlds[(i >> 7) * 129 + (i & 127)] = Wt[(size_t)m0 * C_ + i];
#endif
  __syncthreads();

  int t  = (bw0 + (threadIdx.x >> 5)) % (Mt * Nt);
  int n0 = (t % Nt) * 16;
  const float* Xb = X + (size_t)b * C_ * S_;
  int lane = threadIdx.x & 31, half = lane >> 4, l = lane & 15;
  v8f acc = {};
  for (int k0 = 0; k0 < C_; k0 += 4) {
    int ka = k0 + 2 * half;
    v2f a, bf;
    a.x  = Wlds[l * 129 + ka];
    a.y  = Wlds[l * 129 + ka + 1];
    bf.x = Xb[(size_t)ka * S_ + n0 + l];
    bf.y = Xb[(size_t)(ka + 1) * S_ + n0 + l];
    if (k0 + 4 < C_) __builtin_prefetch(&Xb[(size_t)(k0 + 4 + 2 * half) * S_ + n0 + l], 0, 1);
    acc = wmma_step(acc, a, bf);
  }
  for (int v = 0; v < 8; ++v) {
    int m = m0 + v + 8 * half;
    float r = acc[v] + bias[m];
    if (RES) r += resid[((size_t)b * C_ + m) * S_ + n0 + l];
    Out[((size_t)b * M_ + m) * S_ + n0 + l] = r;
  }
}

// ---------------- Kernel 4: 2x avg-pool of k and v (v written TRANSPOSED: [b,h,m,d]) ----------------
__global__ void __launch_bounds__(256) pool_kv(const float* __restrict__ qkv,
                                               float* __restrict__ kp, float* __restrict__ vp) {
  int i = blockIdx.x * 256 + threadIdx.x;          // exactly 2*B*NH*HD*SP threads
  const int per = B_ * NH_ * HD_ * SP_;
  int which = i / per;                             // 0 -> k, 1 -> v
  int r = i % per;
  int sp = r % SP_;
  int d  = (r / SP_) % HD_;
  int h  = (r / (SP_ * HD_)) % NH_;
  int b  = r / (SP_ * HD_ * NH_);
  int dz = sp % HP_, wy = (sp / HP_) % HP_, hx = sp / (HP_ * HP_);
  const float* src = qkv + ((size_t)b * 3 * C_ + (1 + which) * C_ + h * HD_ + d) * S_;
  float acc = 0.f;
  for (int a = 0; a < 2; ++a)
    for (int w = 0; w < 2; ++w)
      for (int c = 0; c < 2; ++c)
        acc += src[(size_t)(2 * hx + a) * 576 + (2 * wy + w) * 24 + (2 * dz + c)];
  acc *= 0.125f;
  if (which) vp[((size_t)(b * NH_ + h) * SP_ + sp) * HD_ + d] = acc;   // [b,h,m,d]
  else       kp[r] = acc;                                              // [b,h,d,m]
}

// ---------------- Kernel 5: flash attention (16-query tiles, chunked softmax) ----------------
#define CHUNKS  9
#define MTPC    12                 // m-tiles per chunk (3 per wave)
#define CHUNK_M (MTPC * 16)        // 192
#define LDP     196                // padded row stride (mod 64 == 4 -> conflict-free frag loads)

__global__ void __launch_bounds__(128) attn_flash(const float* __restrict__ qkv,
                                                  const float* __restrict__ kp, const float* __restrict__ vp,
                                                  float* __restrict__ attnout) {
  __shared__ float P[16 * LDP];
  __shared__ float pmax[16][8], psum[16][8];
  __shared__ float rmax[16], rsum[16], fbuf[16], nmax[16];
  __shared__ float outpart[4][16][32];

  int blk = blockIdx.x;            // B*NH*(S/16)
  int nt  = blk % (S_ / 16);
  int bh  = blk / (S_ / 16);
  int h = bh % NH_, b = bh / NH_;
  int n0 = nt * 16;

  const float* qbase = qkv + ((size_t)b * 3 * C_ + h * HD_) * S_;
  const float* kbase = kp + ((size_t)(b * NH_ + h) * HD_) * SP_;
  const float* vbase = vp + ((size_t)(b * NH_ + h) * SP_) * HD_;   // transposed [m][d]

  int tid = threadIdx.x;
  int wave = tid >> 5, lane = tid & 31, half = lane >> 4, l = lane & 15;

  if (tid < 16) { rmax[tid] = -1e30f; rsum[tid] = 0.f; }
  v8f acc0 = {}, acc1 = {};        // out accumulators: d 0..15 and 16..31
  __syncthreads();

  for (int ch = 0; ch < CHUNKS; ++ch) {
    int mbase = ch * CHUNK_M;
    // 1) scores S = scale * q^T k  (A = q^T : [n x 32], B = k : [32 x m])
    for (int t = 0; t < 3; ++t) {
      int mt = wave * 3 + t;
      int mb = mbase + mt * 16;
      v8f sacc = {};
      for (int k0 = 0; k0 < HD_; k0 += 4) {
        int ka = k0 + 2 * half;
        v2f a, bf;
        a.x  = qbase[(size_t)ka * S_ + n0 + l];
        a.y  = qbase[(size_t)(ka + 1) * S_ + n0 + l];
        bf.x = kbase[(size_t)ka * SP_ + mb + l];
        bf.y = kbase[(size_t)(ka + 1) * SP_ + mb + l];
        sacc = wmma_step(sacc, a, bf);
      }
      for (int v = 0; v < 8; ++v)
        P[(v + 8 * half) * LDP + mt * 16 + l] = sacc[v] * SCALE_;
    }
    __syncthreads();
    // 2) per-row chunk max partials
    { int r = tid >> 3, cg = tid & 7;
      float m = -1e30f;
      for (int c = cg; c < CHUNK_M; c += 8) m = fmaxf(m, P[r * LDP + c]);
      pmax[r][cg] = m; }
    __syncthreads();
    // 3) running max + rescale factor
    if (tid < 16) {
      float cm = pmax[tid][0];
      for (int j = 1; j < 8; ++j) cm = fmaxf(cm, pmax[tid][j]);
      float nm = fmaxf(rmax[tid], cm);
      fbuf[tid] = __expf(rmax[tid] - nm);
      nmax[tid] = nm;
      rmax[tid] = nm;
    }
    __syncthreads();
    // 4) exponentiate in LDS + partial row sums; rescale output accumulators
    { int r = tid >> 3, cg = tid & 7;
      float nm = nmax[r], s = 0.f;
      for (int c = cg; c < CHUNK_M; c += 8) {
        float p = __expf(P[r * LDP + c] - nm);
        P[r * LDP + c] = p;
        s += p;
      }
      psum[r][cg] = s; }
    for (int v = 0; v < 8; ++v) {
      float f = fbuf[v + 8 * half];
      acc0[v] *= f; acc1[v] *= f;
    }
    __syncthreads();
    if (tid < 16) {
      float s = 0.f;
      for (int j = 0; j < 8; ++j) s += psum[tid][j];
      rsum[tid] = rsum[tid] * fbuf[tid] + s;
    }
    // 5) out += P @ v over this wave's K-slice (A = P[n x m], B = vT[m x d], lane-coalesced)
    for (int t = 0; t < 3; ++t) {
      int mt = wave * 3 + t;
      int mb = mbase + mt * 16;
      for (int k0 = 0; k0 < 16; k0 += 4) {
        int ka = k0 + 2 * half;
        v2f a, b0, b1;
        a.x  = P[l * LDP + mt * 16 + ka];
        a.y  = P[l * LDP + mt * 16 + ka + 1];
        b0.x = vbase[(size_t)(mb + ka) * HD_ + l];
        b0.y = vbase[(size_t)(mb + ka + 1) * HD_ + l];
        b1.x = vbase[(size_t)(mb + ka) * HD_ + 16 + l];
        b1.y = vbase[(size_t)(mb + ka + 1) * HD_ + 16 + l];
        acc0 = wmma_step(acc0, a, b0);
        acc1 = wmma_step(acc1, a, b1);
      }
    }
    __syncthreads();
  }
  // cross-wave reduction + 1/rowsum, write [C,S]-layout for the proj GEMM
  for (int v = 0; v < 8; ++v) {
    outpart[wave][v + 8 * half][l]      = acc0[v];
    outpart[wave][v + 8 * half][16 + l] = acc1[v];
  }
  __syncthreads();
  for (int i = tid; i < 512; i += 128) {
    int r = i >> 5, d = i & 31;
    float s = outpart[0][r][d] + outpart[1][r][d] + outpart[2][r][d] + outpart[3][r][d];
    attnout[((size_t)b * C_ + h * HD_ + d) * S_ + n0 + r] = s / rsum[r];
  }
}

// ---------------- launch ----------------
extern "C" void kernel_launch(void* const* d_in, const int* in_sizes, int n_in,
                              void* d_out, int out_size, void* d_ws, size_t ws_size,
                              hipStream_t stream) {
  const float* x      = (const float*)d_in[0];
  const float* gn_w   = (const float*)d_in[1];
  const float* gn_b   = (const float*)d_in[2];
  const float* qkv_w  = (const float*)d_in[3];
  const float* qkv_b  = (const float*)d_in[4];
  const float* proj_w = (const float*)d_in[5];
  const float* proj_b = (const float*)d_in[6];
  float* out = (float*)d_out;

  float* ws      = (float*)d_ws;
  float* stats   = ws;                                   // 32 (pad 64)
  float* xn      = ws + 64;                              // B*C*S
  float* qkv     = xn + (size_t)B_ * C_ * S_;            // B*3C*S
  float* kpool   = qkv + (size_t)B_ * 3 * C_ * S_;       // B*NH*HD*SP
  float* vpool   = kpool + (size_t)B_ * NH_ * HD_ * SP_; // B*NH*SP*HD (transposed)
  float* attnout = vpool + (size_t)B_ * NH_ * HD_ * SP_; // B*C*S

  hipLaunchKernelGGL(gn_stats, dim3(B_ * G_), dim3(256), 0, stream, x, stats);
  hipLaunchKernelGGL(gn_apply, dim3((B_ * C_ * S_) / 256), dim3(256), 0, stream,
                     x, gn_w, gn_b, stats, xn);
  hipLaunchKernelGGL((gemm_wmma<3 * C_, false>), dim3(B_ * 24 * 864 / 8), dim3(256), 0, stream,
                     xn, qkv_w, qkv_b, (const float*)nullptr, qkv);
  hipLaunchKernelGGL(pool_kv, dim3((2 * B_ * NH_ * HD_ * SP_) / 256), dim3(256), 0, stream,
                     qkv, kpool, vpool);
  hipLaunchKernelGGL(attn_flash, dim3(B_ * NH_ * (S_ / 16)), dim3(128), 0, stream,
                     qkv, kpool, vpool, attnout);
  hipLaunchKernelGGL((gemm_wmma<C_, true>), dim3(B_ * 8 * 864 / 8), dim3(256), 0, stream,
                     attnout, proj_w, proj_b, x, out);
}